// GATConvLayer_87308095193813
// MI455X (gfx1250) — compile-verified
//
#include <hip/hip_runtime.h>
#include <math.h>

typedef float v2f __attribute__((ext_vector_type(2)));
typedef float v8f __attribute__((ext_vector_type(8)));

#define WAVE 32
#define NEG_SLOPE 0.2f

// ---------------------------------------------------------------------------
// GEMM: C[M x Nout] = A[M x K] * W[K x Nout], f32 via V_WMMA_F32_16X16X4_F32.
// One wave computes a 16 x (NT*16) row-strip: NT accumulators, A loaded once.
// W is staged per-block into LDS in the exact WMMA B-operand lane layout,
// zero-padded for ragged Nout.  K (=KK) is a compile-time constant and the
// k-loop is software-pipelined: next step's A (global) and NT B operands (LDS)
// are prefetched into ping-pong registers while the current step's NT WMMAs
// execute, so LDS/VMEM latency is hidden behind the MAC chain instead of
// stalling it.
// LDS image: ldsW[kq][ct][lane] = {W[kq*4+2*half][ct*16+l16],
//                                  W[kq*4+2*half+1][ct*16+l16]}
// (A 16x4 f32 layout: lanes 0-15 hold {K0,K1}, lanes 16-31 hold {K2,K3};
//  C/D: VGPR j -> rows j (lanes 0-15) and j+8 (lanes 16-31)).
// M must be a multiple of 16 (100000 = 6250*16).
// ---------------------------------------------------------------------------
template <int NT, int KK>
__global__ void gat_gemm_wmma_f32(const float* __restrict__ A,
                                  const float* __restrict__ W,
                                  float* __restrict__ C,
                                  int M, int Nout)
{
    extern __shared__ v2f ldsW[];           // (KK/4) * NT * 32 entries
    const int tid = threadIdx.x;

    // ---- stage W into LDS (once per block, W is L2-resident) ----
    const int nEntries = (KK >> 2) * NT * 32;
    for (int i = tid; i < nEntries; i += blockDim.x) {
        int kq   = i / (NT * 32);
        int rem  = i - kq * (NT * 32);
        int ct   = rem >> 5;
        int lane = rem & 31;
        int half = lane >> 4;
        int l16  = lane & 15;
        int ka   = kq * 4 + 2 * half;
        int col  = ct * 16 + l16;
        v2f v = {0.0f, 0.0f};
        if (col < Nout) {
            v.x = W[(size_t)ka * Nout + col];
            v.y = W[(size_t)(ka + 1) * Nout + col];
        }
        ldsW[i] = v;
    }
    __syncthreads();

    const int wave = blockIdx.x * (blockDim.x / WAVE) + (tid / WAVE);
    const int lane = tid & (WAVE - 1);
    const int rowBase = wave * 16;
    if (rowBase >= M) return;

    const int half = lane >> 4;
    const int l16  = lane & 15;
    const float* Arow = A + (size_t)(rowBase + l16) * KK + 2 * half;

    v8f acc[NT];
#pragma unroll
    for (int t = 0; t < NT; ++t) acc[t] = (v8f){};

    // ---- software-pipelined MAC loop (prologue: load step 0) ----
    const v2f* wp = ldsW + lane;
    v2f a_cur, b_cur[NT];
    a_cur.x = Arow[0];
    a_cur.y = Arow[1];
#pragma unroll
    for (int t = 0; t < NT; ++t) b_cur[t] = wp[t * 32];

#pragma unroll 2
    for (int k0 = 0; k0 < KK - 4; k0 += 4) {
        // prefetch next step while current WMMAs run
        v2f a_nxt, b_nxt[NT];
        a_nxt.x = Arow[k0 + 4];
        a_nxt.y = Arow[k0 + 5];
        const v2f* wpn = wp + NT * 32;
#pragma unroll
        for (int t = 0; t < NT; ++t) b_nxt[t] = wpn[t * 32];

#pragma unroll
        for (int t = 0; t < NT; ++t)
            acc[t] = __builtin_amdgcn_wmma_f32_16x16x4_f32(
                         false, a_cur, false, b_cur[t], (short)0, acc[t],
                         false, false);

        a_cur = a_nxt;
#pragma unroll
        for (int t = 0; t < NT; ++t) b_cur[t] = b_nxt[t];
        wp = wpn;
    }
    // epilogue: last k-step
#pragma unroll
    for (int t = 0; t < NT; ++t)
        acc[t] = __builtin_amdgcn_wmma_f32_16x16x4_f32(
                     false, a_cur, false, b_cur[t], (short)0, acc[t],
                     false, false);

#pragma unroll
    for (int t = 0; t < NT; ++t) {
        int bcol = t * 16 + l16;
        if (bcol < Nout) {
#pragma unroll
            for (int j = 0; j < 8; ++j) {
                int r = rowBase + j + 8 * half;
                C[(size_t)r * Nout + bcol] = acc[t][j];
            }
        }
    }
}

// el[n,h] = <feat[n,h,:], attn_l[h,:]>, er likewise. One thread per (n,h).
__global__ void gat_node_dots(const float* __restrict__ feat,
                              const float* __restrict__ attn_l,
                              const float* __restrict__ attn_r,
                              float* __restrict__ el, float* __restrict__ er,
                              int Nn, int H, int F)
{
    int gid = blockIdx.x * blockDim.x + threadIdx.x;
    if (gid >= Nn * H) return;
    int h = gid % H;
    const float* f  = feat + (size_t)gid * F;
    const float* al = attn_l + (size_t)h * F;
    const float* ar = attn_r + (size_t)h * F;
    float sl = 0.0f, sr = 0.0f;
    for (int i = 0; i < F; ++i) { float v = f[i]; sl += v * al[i]; sr += v * ar[i]; }
    el[gid] = sl;
    er[gid] = sr;
}

__global__ void gat_fill_f32(float* __restrict__ p, float v, int n)
{
    int i = blockIdx.x * blockDim.x + threadIdx.x;
    if (i < n) p[i] = v;
}

// Float atomic max via int/uint ordering trick (init must be -inf).
__device__ __forceinline__ void atomicMaxF32(float* addr, float val)
{
    if (val >= 0.0f) atomicMax((int*)addr, __float_as_int(val));
    else             atomicMin((unsigned int*)addr, __float_as_uint(val));
}

// Pass 1: e = leaky_relu(el[src] + er[dst]); emax[dst] = max(...)
__global__ void gat_edge_logits_max(const int* __restrict__ src,
                                    const int* __restrict__ dst,
                                    const float* __restrict__ el,
                                    const float* __restrict__ er,
                                    float* __restrict__ eb,
                                    float* __restrict__ emax,
                                    int E, int H)
{
    int gid = blockIdx.x * blockDim.x + threadIdx.x;
    if (gid >= E * H) return;
    int eid = gid / H;
    int h   = gid - eid * H;
    int s = src[eid], d = dst[eid];
    float v = el[s * H + h] + er[d * H + h];
    v = (v > 0.0f) ? v : NEG_SLOPE * v;
    eb[gid] = v;
    atomicMaxF32(&emax[d * H + h], v);
}

// Pass 2: ee = exp(e - emax[dst]); denom[dst] += ee (eb overwritten with ee)
__global__ void gat_edge_exp_sum(const int* __restrict__ dst,
                                 float* __restrict__ eb,
                                 const float* __restrict__ emax,
                                 float* __restrict__ denom,
                                 int E, int H)
{
    int gid = blockIdx.x * blockDim.x + threadIdx.x;
    if (gid >= E * H) return;
    int eid = gid / H;
    int h   = gid - eid * H;
    int d = dst[eid];
    float ee = __expf(eb[gid] - emax[d * H + h]);
    eb[gid] = ee;
    atomicAdd(&denom[d * H + h], ee);
}

// Pass 2.5: alpha = ee / denom[dst]  (eb overwritten with alpha) so the big
// scatter pass does no per-feature denom gathers or divisions.
__global__ void gat_edge_alpha(const int* __restrict__ dst,
                               float* __restrict__ eb,
                               const float* __restrict__ denom,
                               int E, int H)
{
    int gid = blockIdx.x * blockDim.x + threadIdx.x;
    if (gid >= E * H) return;
    int eid = gid / H;
    int h   = gid - eid * H;
    int d = dst[eid];
    float dn = denom[d * H + h];
    eb[gid] = eb[gid] / (dn > 0.0f ? dn : 1.0f);
}

// Pass 3: rst[dst, c] += alpha(edge,h) * feat[src, c].  Thread per (edge, c),
// c in [0, H*F) so gathers and atomic adds are coalesced along features.
__global__ void gat_edge_scatter(const int* __restrict__ src,
                                 const int* __restrict__ dst,
                                 const float* __restrict__ eb,
                                 const float* __restrict__ feat,
                                 float* __restrict__ rst,
                                 int E, int H, int F)
{
    int HF = H * F;
    int gid = blockIdx.x * blockDim.x + threadIdx.x;
    if (gid >= E * HF) return;
    int eid = gid / HF;
    int c   = gid - eid * HF;
    int h   = c / F;
    int s = src[eid], d = dst[eid];
    float alpha = eb[eid * H + h];
    atomicAdd(&rst[(size_t)d * HF + c], alpha * feat[(size_t)s * HF + c]);
}

// out[n, c] = (rst[n, c] + bias[c]), optional ReLU
__global__ void gat_bias_act(const float* __restrict__ rst,
                             const float* __restrict__ bias,
                             float* __restrict__ out,
                             int total, int C, int doRelu)
{
    int gid = blockIdx.x * blockDim.x + threadIdx.x;
    if (gid >= total) return;
    int c = gid % C;
    float v = rst[gid] + bias[c];
    if (doRelu) v = v > 0.0f ? v : 0.0f;
    out[gid] = v;
}

static inline int ceil_div(long long a, long long b) { return (int)((a + b - 1) / b); }

extern "C" void kernel_launch(void* const* d_in, const int* in_sizes, int n_in,
                              void* d_out, int out_size, void* d_ws, size_t ws_size,
                              hipStream_t stream)
{
    const float* in_feat  = (const float*)d_in[0];
    const float* W1       = (const float*)d_in[1];
    const float* attn_l1  = (const float*)d_in[2];
    const float* attn_r1  = (const float*)d_in[3];
    const float* b1       = (const float*)d_in[4];
    const float* W2       = (const float*)d_in[5];
    const float* attn_l2  = (const float*)d_in[6];
    const float* attn_r2  = (const float*)d_in[7];
    const float* b2       = (const float*)d_in[8];
    const int*   src      = (const int*)d_in[9];
    const int*   dst      = (const int*)d_in[10];

    const int IN_F = 128;
    const int H1 = 2, F1 = 64, HF1 = H1 * F1;   // 128
    const int H2 = 1, F2 = 40, HF2 = H2 * F2;   // 40
    constexpr int NT1 = 8;                      // 128/16 col tiles
    constexpr int NT2 = 3;                      // ceil(40/16) col tiles (ragged)
    constexpr int K1 = 128;                     // layer1 reduction dim
    constexpr int K2 = 128;                     // layer2 reduction dim
    const int Nn = in_sizes[0] / IN_F;          // 100000
    const int E  = in_sizes[9];                 // 1600000

    // ---- workspace carve-out (float units, 256-elt aligned) ----
    float* ws = (float*)d_ws;
    size_t off = 0;
    auto carve = [&](size_t nfl) -> float* {
        float* p = ws + off;
        off += (nfl + 255) & ~(size_t)255;
        return p;
    };
    float* featA = carve((size_t)Nn * HF1); // layer1 feat, later relu'd hidden
    float* featB = carve((size_t)Nn * HF1); // layer1 rst, later layer2 feat
    float* rst2  = carve((size_t)Nn * HF2); // layer2 aggregation
    float* el    = carve((size_t)Nn * H1);
    float* er    = carve((size_t)Nn * H1);
    float* emax  = carve((size_t)Nn * H1);
    float* denom = carve((size_t)Nn * H1);
    float* eb    = carve((size_t)E  * H1);  // logits -> exp -> alpha

    const int BLK = 256;
    const int wavesPerBlk = BLK / WAVE;
    const int rowWaves = (Nn + 15) / 16;    // one wave per 16-row strip

    // =============== Layer 1 ===============
    {
        size_t ldsBytes = (size_t)(K1 / 4) * NT1 * 32 * sizeof(v2f); // 64 KB
        gat_gemm_wmma_f32<NT1, K1><<<ceil_div(rowWaves, wavesPerBlk), BLK, ldsBytes, stream>>>(
            in_feat, W1, featA, Nn, HF1);

        gat_node_dots<<<ceil_div((long long)Nn * H1, BLK), BLK, 0, stream>>>(
            featA, attn_l1, attn_r1, el, er, Nn, H1, F1);

        hipMemsetAsync(denom, 0, (size_t)Nn * H1 * sizeof(float), stream);
        hipMemsetAsync(featB, 0, (size_t)Nn * HF1 * sizeof(float), stream);
        gat_fill_f32<<<ceil_div((long long)Nn * H1, BLK), BLK, 0, stream>>>(
            emax, -INFINITY, Nn * H1);

        gat_edge_logits_max<<<ceil_div((long long)E * H1, BLK), BLK, 0, stream>>>(
            src, dst, el, er, eb, emax, E, H1);
        gat_edge_exp_sum<<<ceil_div((long long)E * H1, BLK), BLK, 0, stream>>>(
            dst, eb, emax, denom, E, H1);
        gat_edge_alpha<<<ceil_div((long long)E * H1, BLK), BLK, 0, stream>>>(
            dst, eb, denom, E, H1);
        gat_edge_scatter<<<ceil_div((long long)E * HF1, BLK), BLK, 0, stream>>>(
            src, dst, eb, featA, featB, E, H1, F1);

        // h = relu(rst1 + b1)  (featB -> featA)
        gat_bias_act<<<ceil_div((long long)Nn * HF1, BLK), BLK, 0, stream>>>(
            featB, b1, featA, Nn * HF1, HF1, 1);
    }

    // =============== Layer 2 ===============
    {
        size_t ldsBytes = (size_t)(K2 / 4) * NT2 * 32 * sizeof(v2f); // 24 KB
        gat_gemm_wmma_f32<NT2, K2><<<ceil_div(rowWaves, wavesPerBlk), BLK, ldsBytes, stream>>>(
            featA, W2, featB, Nn, HF2);

        gat_node_dots<<<ceil_div((long long)Nn * H2, BLK), BLK, 0, stream>>>(
            featB, attn_l2, attn_r2, el, er, Nn, H2, F2);

        hipMemsetAsync(denom, 0, (size_t)Nn * H2 * sizeof(float), stream);
        hipMemsetAsync(rst2,  0, (size_t)Nn * HF2 * sizeof(float), stream);
        gat_fill_f32<<<ceil_div((long long)Nn * H2, BLK), BLK, 0, stream>>>(
            emax, -INFINITY, Nn * H2);

        gat_edge_logits_max<<<ceil_div((long long)E * H2, BLK), BLK, 0, stream>>>(
            src, dst, el, er, eb, emax, E, H2);
        gat_edge_exp_sum<<<ceil_div((long long)E * H2, BLK), BLK, 0, stream>>>(
            dst, eb, emax, denom, E, H2);
        gat_edge_alpha<<<ceil_div((long long)E * H2, BLK), BLK, 0, stream>>>(
            dst, eb, denom, E, H2);
        gat_edge_scatter<<<ceil_div((long long)E * HF2, BLK), BLK, 0, stream>>>(
            src, dst, eb, featB, rst2, E, H2, F2);

        // out = rst2 + b2 (no relu)
        gat_bias_act<<<ceil_div((long long)Nn * HF2, BLK), BLK, 0, stream>>>(
            rst2, b2, (float*)d_out, Nn * HF2, HF2, 0);
    }
}